// RPNtoRoI_11742440588153
// MI455X (gfx1250) — compile-verified
//
#include <hip/hip_runtime.h>
#include <hip/hip_fp16.h>

#define B_        4
#define N_ANCH    36864
#define NMS_TOPN  300
#define TOTAL_POS 64
#define TOTAL_BB  128
#define NLAB      21
#define FH        64
#define FW        64
#define FC        512
#define NEG_INF_F (-1.0e9f)

#define TPB 512
#define EPT (N_ANCH / TPB) /* 72 */

typedef float v2f __attribute__((ext_vector_type(2)));
typedef float v8f __attribute__((ext_vector_type(8)));

static __device__ __forceinline__ unsigned pk2(float a, float b) {
  __half2 h = __floats2half2_rn(a, b);
  unsigned r;
  __builtin_memcpy(&r, &h, 4);
  return r;
}
static __device__ __forceinline__ float2 up2(unsigned u) {
  __half2 h;
  __builtin_memcpy(&h, &u, 4);
  return __half22float2(h);
}
// Monotonic key: higher score wins, ties -> smaller index wins (== jnp.argmax).
static __device__ __forceinline__ unsigned long long mkkey(float s, unsigned g) {
  unsigned u = __float_as_uint(s);
  u = (u & 0x80000000u) ? ~u : (u | 0x80000000u);
  return ((unsigned long long)u << 32) | (unsigned long long)(0xFFFFFFFFu - g);
}
static __device__ __forceinline__ float keyscore(unsigned long long k) {
  unsigned v = (unsigned)(k >> 32);
  return (v & 0x80000000u) ? __uint_as_float(v & 0x7FFFFFFFu) : __uint_as_float(~v);
}

// ---------------------------------------------------------------------------
// Kernel 1: decode_boxes + iterative NMS (one workgroup per batch).
// Scores live in registers (EPT per thread), boxes cached as packed fp16 in ws.
// ---------------------------------------------------------------------------
__global__ __launch_bounds__(TPB) void nms_kernel(
    const float* __restrict__ rpn_deltas, const float* __restrict__ rpn_scores,
    const float* __restrict__ anchors, float* __restrict__ ws_boxes,
    unsigned* __restrict__ ws_hbox, float* __restrict__ ws_nms) {
  __shared__ unsigned long long red[TPB];
  __shared__ unsigned selArr[NMS_TOPN];

  const int b = blockIdx.x;
  const int tid = threadIdx.x;
  const float* ab = anchors + (size_t)b * N_ANCH * 4;
  const float* db = rpn_deltas + (size_t)b * N_ANCH * 4;
  const float* sb = rpn_scores + (size_t)b * N_ANCH;
  float* boxb = ws_boxes + (size_t)b * N_ANCH * 4;
  unsigned* hb = ws_hbox + (size_t)b * N_ANCH * 2;

  float sc[EPT];

  // ---- decode ----
#pragma unroll
  for (int e = 0; e < EPT; ++e) {
    int g = e * TPB + tid;
    float a0 = ab[g * 4 + 0], a1 = ab[g * 4 + 1];
    float a2 = ab[g * 4 + 2], a3 = ab[g * 4 + 3];
    float d0 = db[g * 4 + 0], d1 = db[g * 4 + 1];
    float d2 = db[g * 4 + 2], d3 = db[g * 4 + 3];
    float aw = a3 - a1, ah = a2 - a0;
    float acx = a1 + 0.5f * aw, acy = a0 + 0.5f * ah;
    float w = expf(d3) * aw, h = expf(d2) * ah;
    float cx = d1 * aw + acx, cy = d0 * ah + acy;
    float y1 = cy - 0.5f * h, x1 = cx - 0.5f * w;
    float y2 = y1 + h, x2 = x1 + w;
    boxb[g * 4 + 0] = y1; boxb[g * 4 + 1] = x1;
    boxb[g * 4 + 2] = y2; boxb[g * 4 + 3] = x2;
    hb[g * 2 + 0] = pk2(y1, x1);
    hb[g * 2 + 1] = pk2(y2, x2);
    sc[e] = sb[g];
  }
  __threadfence_block();
  __syncthreads();

  // ---- initial local argmax ----
  unsigned long long best = 0ull;
#pragma unroll
  for (int e = 0; e < EPT; ++e) {
    unsigned long long k = mkkey(sc[e], (unsigned)(e * TPB + tid));
    best = (k > best) ? k : best;
  }

#pragma unroll 1
  for (int it = 0; it < NMS_TOPN; ++it) {
    red[tid] = best;
    __syncthreads();
    for (int s = TPB / 2; s > 0; s >>= 1) {
      if (tid < s) {
        unsigned long long o = red[tid + s];
        if (o > red[tid]) red[tid] = o;
      }
      __syncthreads();
    }
    unsigned long long win = red[0];
    __syncthreads();

    unsigned selIdx = 0xFFFFFFFFu - (unsigned)(win & 0xFFFFFFFFull);
    float selScore = keyscore(win);
    int valid = selScore > NEG_INF_F * 0.5f;
    if (tid == 0) selArr[it] = valid ? selIdx : 0xFFFFFFFFu;

    float2 p0 = up2(hb[selIdx * 2 + 0]);
    float2 p1 = up2(hb[selIdx * 2 + 1]);
    float qy1 = p0.x, qx1 = p0.y, qy2 = p1.x, qx2 = p1.y;
    float qarea = (qy2 - qy1) * (qx2 - qx1);

    best = 0ull;
#pragma unroll
    for (int e = 0; e < EPT; ++e) {
      unsigned g = (unsigned)(e * TPB + tid);
      float2 c0 = up2(hb[g * 2 + 0]);
      float2 c1 = up2(hb[g * 2 + 1]);
      float yt = fmaxf(qy1, c0.x), xt = fmaxf(qx1, c0.y);
      float yb = fminf(qy2, c1.x), xb = fminf(qx2, c1.y);
      float inter = fmaxf(yb - yt, 0.0f) * fmaxf(xb - xt, 0.0f);
      float carea = (c1.x - c0.x) * (c1.y - c0.y);
      float uni = qarea + carea - inter;
      bool supp = ((uni > 0.0f) && (inter > 0.5f * uni)) || (g == selIdx);
      sc[e] = supp ? NEG_INF_F : sc[e];
      unsigned long long k = mkkey(sc[e], g);
      best = (k > best) ? k : best;
    }
  }
  __syncthreads();

  if (tid < NMS_TOPN) {
    unsigned gi = selArr[tid];
    float o0 = 0.f, o1 = 0.f, o2 = 0.f, o3 = 0.f;
    if (gi != 0xFFFFFFFFu) {
      o0 = fminf(fmaxf(boxb[gi * 4 + 0], 0.f), 1.f);
      o1 = fminf(fmaxf(boxb[gi * 4 + 1], 0.f), 1.f);
      o2 = fminf(fmaxf(boxb[gi * 4 + 2], 0.f), 1.f);
      o3 = fminf(fmaxf(boxb[gi * 4 + 3], 0.f), 1.f);
    }
    float* o = ws_nms + ((size_t)b * NMS_TOPN + tid) * 4;
    o[0] = o0; o[1] = o1; o[2] = o2; o[3] = o3;
  }
}

// ---------------------------------------------------------------------------
// Kernel 2: IoU vs GT, stable descending rank of 300, pick 64 pos + 64 neg,
// deltas + one-hot epilogues (one workgroup per batch).
// ---------------------------------------------------------------------------
__global__ __launch_bounds__(512) void select_kernel(
    const float* __restrict__ ws_nms, const float* __restrict__ gt_boxes,
    const int* __restrict__ gt_labels, float* __restrict__ ws_roi,
    float* __restrict__ out, unsigned long long deltas_off,
    unsigned long long labels_off) {
  __shared__ float nb[NMS_TOPN][4];
  __shared__ float gtb[16][4];
  __shared__ int gtl[16];
  __shared__ float merged[NMS_TOPN];
  __shared__ int maxg[NMS_TOPN];
  __shared__ int selA[TOTAL_BB];

  int b = blockIdx.x, tid = threadIdx.x;
  if (tid < 16) {
    gtb[tid][0] = gt_boxes[((size_t)b * 16 + tid) * 4 + 0];
    gtb[tid][1] = gt_boxes[((size_t)b * 16 + tid) * 4 + 1];
    gtb[tid][2] = gt_boxes[((size_t)b * 16 + tid) * 4 + 2];
    gtb[tid][3] = gt_boxes[((size_t)b * 16 + tid) * 4 + 3];
    gtl[tid] = gt_labels[(size_t)b * 16 + tid];
  }
  if (tid < NMS_TOPN) {
    const float* p = ws_nms + ((size_t)b * NMS_TOPN + tid) * 4;
    nb[tid][0] = p[0]; nb[tid][1] = p[1]; nb[tid][2] = p[2]; nb[tid][3] = p[3];
  }
  __syncthreads();

  if (tid < NMS_TOPN) {
    float by1 = nb[tid][0], bx1 = nb[tid][1], by2 = nb[tid][2], bx2 = nb[tid][3];
    float ba = (by2 - by1) * (bx2 - bx1);
    float bestv = -1.f;
    int bestj = 0;
    for (int j = 0; j < 16; ++j) {
      float yt = fmaxf(by1, gtb[j][0]), xt = fmaxf(bx1, gtb[j][1]);
      float yb = fminf(by2, gtb[j][2]), xb = fminf(bx2, gtb[j][3]);
      float inter = fmaxf(yb - yt, 0.f) * fmaxf(xb - xt, 0.f);
      float ga = (gtb[j][2] - gtb[j][0]) * (gtb[j][3] - gtb[j][1]);
      float uni = ba + ga - inter;
      float iou = (uni > 0.f) ? (inter / uni) : 0.f;
      if (iou > bestv) { bestv = iou; bestj = j; }
    }
    merged[tid] = bestv;
    maxg[tid] = bestj;
  }
  __syncthreads();

  if (tid < NMS_TOPN) {
    float mi = merged[tid];
    int r = 0;
    for (int j = 0; j < NMS_TOPN; ++j) {
      float mj = merged[j];
      r += (mj > mi) || ((mj == mi) && (j < tid));
    }
    if (r < TOTAL_BB) selA[r] = tid;
  }
  __syncthreads();

  if (tid < TOTAL_BB) {
    int i = selA[tid];
    float by1 = nb[i][0], bx1 = nb[i][1], by2 = nb[i][2], bx2 = nb[i][3];
    float* roi = ws_roi + ((size_t)b * TOTAL_BB + tid) * 4;
    roi[0] = by1; roi[1] = bx1; roi[2] = by2; roi[3] = bx2;

    float g0, g1, g2, g3;
    int lab;
    if (tid < TOTAL_POS) {
      int gi = maxg[i];
      g0 = gtb[gi][0]; g1 = gtb[gi][1]; g2 = gtb[gi][2]; g3 = gtb[gi][3];
      lab = gtl[gi];
    } else {
      g0 = g1 = g2 = g3 = 0.f;
      lab = NLAB - 1;
    }
    float bw = bx2 - bx1, bh = by2 - by1;
    float bcx = bx1 + 0.5f * bw, bcy = by1 + 0.5f * bh;
    float gw = g3 - g1, gh = g2 - g0;
    float gcx = g1 + 0.5f * gw, gcy = g0 + 0.5f * gh;
    bw = (bw == 0.f) ? 1e-3f : bw;
    bh = (bh == 0.f) ? 1e-3f : bh;
    float dx = (gw == 0.f) ? 0.f : (gcx - bcx) / bw;
    float dy = (gh == 0.f) ? 0.f : (gcy - bcy) / bh;
    float dw = (gw == 0.f) ? 0.f : logf(fmaxf(gw, 1e-12f) / bw);
    float dh = (gh == 0.f) ? 0.f : logf(fmaxf(gh, 1e-12f) / bh);
    float dd[4] = {dy, dx, dh, dw};

    float* od = out + deltas_off + ((size_t)b * TOTAL_BB + tid) * (NLAB * 4);
    for (int l = 0; l < NLAB; ++l)
      for (int k = 0; k < 4; ++k) od[l * 4 + k] = (l == lab) ? dd[k] : 0.f;
    float* ol = out + labels_off + ((size_t)b * TOTAL_BB + tid) * NLAB;
    for (int l = 0; l < NLAB; ++l) ol[l] = (l == lab) ? 1.f : 0.f;
  }
}

// ---------------------------------------------------------------------------
// Kernel 3: RoI-align via separable bilinear; x-interp as a 7x14 x 14x512
// GEMM on the matrix pipe (V_WMMA_F32_16X16X4_F32). One block per (b,roi,py).
// ---------------------------------------------------------------------------
#define MSTRIDE 520 /* 512 + 8: kills the 2-way bank conflict on B-frag loads */

__global__ __launch_bounds__(256) void pool_kernel(
    const float* __restrict__ fm, const float* __restrict__ ws_roi,
    float* __restrict__ out) {
  __shared__ float Wx[16][16];
  __shared__ float Ml[16][MSTRIDE];
  __shared__ int xIdx[16];

  int blk = blockIdx.x;
  int py = blk % 7;
  int roi = (blk / 7) % TOTAL_BB;
  int b = blk / (7 * TOTAL_BB);
  int tid = threadIdx.x;

  const float* box = ws_roi + ((size_t)b * TOTAL_BB + roi) * 4;
  float y1 = box[0], x1 = box[1], y2 = box[2], x2 = box[3];

  float iy = (float)py / 6.0f;
  float ys = y1 * 63.f + iy * ((y2 - y1) * 63.f);
  float y0f = floorf(ys);
  float wy = ys - y0f;
  int y0i = min(max((int)y0f, 0), FH - 1);
  int y1i = min(y0i + 1, FH - 1);
  float vy = (ys >= 0.f && ys <= (float)(FH - 1)) ? 1.f : 0.f;

  { // Wx weight matrix (rows 7..15 and cols 14..15 zero; vx folded in)
    int px = tid >> 4, k = tid & 15;
    float v = 0.f;
    if (px < 7) {
      float ix = (float)px / 6.0f;
      float xs = x1 * 63.f + ix * ((x2 - x1) * 63.f);
      float x0f = floorf(xs);
      float wx = xs - x0f;
      float vx = (xs >= 0.f && xs <= (float)(FW - 1)) ? 1.f : 0.f;
      if (k == 2 * px) v = vx * (1.f - wx);
      else if (k == 2 * px + 1) v = vx * wx;
    }
    Wx[px][k] = v;
  }
  if (tid < 16) {
    int xv = 0;
    if (tid < 14) {
      int px = tid >> 1;
      float ix = (float)px / 6.0f;
      float xs = x1 * 63.f + ix * ((x2 - x1) * 63.f);
      int x0i = min(max((int)floorf(xs), 0), FW - 1);
      int x1ii = min(x0i + 1, FW - 1);
      xv = (tid & 1) ? x1ii : x0i;
    }
    xIdx[tid] = xv;
  }
  __syncthreads();

  // y-blend the 14 needed columns into the K x C panel
  const float* fb = fm + (size_t)b * FH * FW * FC;
  const float* rowT = fb + (size_t)y0i * FW * FC;
  const float* rowB = fb + (size_t)y1i * FW * FC;
  for (int idx = tid; idx < 16 * FC; idx += 256) {
    int k = idx >> 9, c = idx & (FC - 1);
    float v = 0.f;
    if (k < 14) {
      int x = xIdx[k];
      float t = rowT[(size_t)x * FC + c];
      float bt = rowB[(size_t)x * FC + c];
      v = (1.f - wy) * t + wy * bt;
    }
    Ml[k][c] = v;
  }
  __syncthreads();

  int wid = tid >> 5, lane = tid & 31;
  int m = lane & 15;
  int hi = (lane >> 4) << 1; // K offset 0 (lanes 0-15) or 2 (lanes 16-31)

  // A fragments (16x4 slices of Wx), per documented f32 A layout:
  // VGPR0: K = kc*4 + {0|2}, VGPR1: K = kc*4 + {1|3}
  v2f afr[4];
#pragma unroll
  for (int kc = 0; kc < 4; ++kc) {
    afr[kc].x = Wx[m][kc * 4 + hi];
    afr[kc].y = Wx[m][kc * 4 + hi + 1];
  }

  unsigned long long obase =
      ((((unsigned long long)b * TOTAL_BB + roi) * 7 + py) * 7) * FC;

  for (int t = wid; t < FC / 16; t += 8) {
    v8f acc = {0.f, 0.f, 0.f, 0.f, 0.f, 0.f, 0.f, 0.f};
#pragma unroll
    for (int kc = 0; kc < 4; ++kc) {
      v2f bfr;
      bfr.x = Ml[kc * 4 + hi][t * 16 + m];
      bfr.y = Ml[kc * 4 + hi + 1][t * 16 + m];
      acc = __builtin_amdgcn_wmma_f32_16x16x4_f32(
          false, afr[kc], false, bfr, (short)0, acc, false, false);
    }
    if (lane < 16) {
#pragma unroll
      for (int r = 0; r < 7; ++r) {
        out[obase + (unsigned long long)r * FC + t * 16 + lane] = vy * acc[r];
      }
    }
  }
}

// ---------------------------------------------------------------------------
extern "C" void kernel_launch(void* const* d_in, const int* in_sizes, int n_in,
                              void* d_out, int out_size, void* d_ws,
                              size_t ws_size, hipStream_t stream) {
  (void)in_sizes; (void)n_in; (void)out_size; (void)ws_size;
  const float* fm = (const float*)d_in[0];
  const float* rpn_d = (const float*)d_in[1];
  const float* rpn_l = (const float*)d_in[2];
  const float* anch = (const float*)d_in[3];
  const float* gtb = (const float*)d_in[4];
  const int* gtl = (const int*)d_in[5];
  float* out = (float*)d_out;

  float* ws_boxes = (float*)d_ws;                                 // B*N*4 f32
  unsigned* ws_hbox = (unsigned*)(ws_boxes + (size_t)B_ * N_ANCH * 4); // B*N*2
  float* ws_nms = (float*)(ws_hbox + (size_t)B_ * N_ANCH * 2);    // B*300*4
  float* ws_roi = ws_nms + (size_t)B_ * NMS_TOPN * 4;             // B*128*4

  const unsigned long long FINAL_ELEMS =
      (unsigned long long)B_ * TOTAL_BB * 7 * 7 * FC;           // 12,845,056
  const unsigned long long DELTAS_OFF = FINAL_ELEMS;
  const unsigned long long LABELS_OFF =
      FINAL_ELEMS + (unsigned long long)B_ * TOTAL_BB * NLAB * 4;

  nms_kernel<<<B_, TPB, 0, stream>>>(rpn_d, rpn_l, anch, ws_boxes, ws_hbox,
                                     ws_nms);
  select_kernel<<<B_, 512, 0, stream>>>(ws_nms, gtb, gtl, ws_roi, out,
                                        DELTAS_OFF, LABELS_OFF);
  pool_kernel<<<B_ * TOTAL_BB * 7, 256, 0, stream>>>(fm, ws_roi, out);
}